// GraphSAGEPredictor_32341103739257
// MI455X (gfx1250) — compile-verified
//
#include <hip/hip_runtime.h>
#include <hip/hip_bf16.h>

typedef __attribute__((ext_vector_type(2))) float v2f;
typedef __attribute__((ext_vector_type(8))) float v8f;

#define WAVES_PER_BLOCK 4
#define S_NBR 10

// ---------------------------------------------------------------------------
// Layer 1: h1[r] = relu( [feat(self_r) | mean_s feat(nbr_s)] @ [Wself1;Wnbr1] + b1 )
// One wave computes a 16-row x 128-col output tile with V_WMMA_F32_16X16X4_F32.
// X tile (16 x 256 f32) staged in LDS; B fragments streamed from global (L2).
// All 8 N-tile accumulators held in registers; A read from LDS once per K step.
// ---------------------------------------------------------------------------
__global__ void sage_layer1(const float* __restrict__ features,
                            const float* __restrict__ Wself1,
                            const float* __restrict__ Wnbr1,
                            const float* __restrict__ b1,
                            const int* __restrict__ input_nodes,
                            const int* __restrict__ nbr1,
                            float* __restrict__ h1,
                            int n1) {
    extern __shared__ float smem[];                       // WAVES_PER_BLOCK * 16 * 256
    const int wave = threadIdx.x >> 5;
    const int lane = threadIdx.x & 31;
    float* X = smem + wave * 16 * 256;
    const int rowBase = (blockIdx.x * WAVES_PER_BLOCK + wave) * 16;

    // ---- stage X: 16 rows, cols [0,128)=self, [128,256)=nbr-mean
    const int c = lane * 4;                               // 32 lanes x float4 = 128 cols
    for (int r = 0; r < 16; ++r) {
        int g  = rowBase + r;
        int gg = (g < n1) ? g : 0;                        // clamp tail (store guarded later)
        const float* selfp = features + (size_t)input_nodes[gg] * 128;
        float4 sv = *(const float4*)(selfp + c);
        float ax = 0.f, ay = 0.f, az = 0.f, aw = 0.f;
        for (int s = 0; s < S_NBR; ++s) {
            int nb = nbr1[(size_t)gg * S_NBR + s];
            const float* fp = features + (size_t)input_nodes[nb] * 128;
            float4 v = *(const float4*)(fp + c);
            ax += v.x; ay += v.y; az += v.z; aw += v.w;
        }
        float* xr = X + r * 256;
        xr[c + 0] = sv.x; xr[c + 1] = sv.y; xr[c + 2] = sv.z; xr[c + 3] = sv.w;
        xr[128 + c + 0] = ax * 0.1f; xr[128 + c + 1] = ay * 0.1f;
        xr[128 + c + 2] = az * 0.1f; xr[128 + c + 3] = aw * 0.1f;
    }
    __syncthreads();

    // ---- WMMA: Y(16x128) = X(16x256) @ [Wself1;Wnbr1](256x128), K stepped by 4
    const int half = lane >> 4;
    const int lo   = lane & 15;
    const float* Xrow = X + lo * 256;

    v8f acc[8];
#pragma unroll
    for (int t = 0; t < 8; ++t) acc[t] = (v8f){0.f, 0.f, 0.f, 0.f, 0.f, 0.f, 0.f, 0.f};

    // self half of K (rows 0..127 -> Wself1)
    {
        const float* wp = Wself1 + (size_t)(2 * half) * 128 + lo;
        for (int k = 0; k < 32; ++k) {
            const int kk = 4 * k + 2 * half;
            v2f a;
            a.x = Xrow[kk];
            a.y = Xrow[kk + 1];
#pragma unroll
            for (int t = 0; t < 8; ++t) {
                v2f b;
                b.x = wp[t * 16];
                b.y = wp[128 + t * 16];
                acc[t] = __builtin_amdgcn_wmma_f32_16x16x4_f32(
                    false, a, false, b, (short)0, acc[t], false, false);
            }
            wp += 4 * 128;
        }
    }
    // neighbor half of K (rows 128..255 -> Wnbr1)
    {
        const float* wp = Wnbr1 + (size_t)(2 * half) * 128 + lo;
        for (int k = 0; k < 32; ++k) {
            const int kk = 128 + 4 * k + 2 * half;
            v2f a;
            a.x = Xrow[kk];
            a.y = Xrow[kk + 1];
#pragma unroll
            for (int t = 0; t < 8; ++t) {
                v2f b;
                b.x = wp[t * 16];
                b.y = wp[128 + t * 16];
                acc[t] = __builtin_amdgcn_wmma_f32_16x16x4_f32(
                    false, a, false, b, (short)0, acc[t], false, false);
            }
            wp += 4 * 128;
        }
    }

#pragma unroll
    for (int t = 0; t < 8; ++t) {
        const int n = t * 16 + lo;
        const float bias = b1[n];
#pragma unroll
        for (int j = 0; j < 8; ++j) {
            int row = rowBase + 8 * half + j;
            if (row < n1) {
                float v = acc[t][j] + bias;
                h1[(size_t)row * 128 + n] = v > 0.f ? v : 0.f;
            }
        }
    }
}

// ---------------------------------------------------------------------------
// Layer 2: h2[r] = [h1[r] | mean_s h1[nbr2[r][s]]] @ [Wself2;Wnbr2] + b2  (64 cols)
// ---------------------------------------------------------------------------
__global__ void sage_layer2(const float* __restrict__ h1,
                            const float* __restrict__ Wself2,
                            const float* __restrict__ Wnbr2,
                            const float* __restrict__ b2,
                            const int* __restrict__ nbr2,
                            float* __restrict__ h2,
                            int n2) {
    extern __shared__ float smem[];
    const int wave = threadIdx.x >> 5;
    const int lane = threadIdx.x & 31;
    float* X = smem + wave * 16 * 256;
    const int rowBase = (blockIdx.x * WAVES_PER_BLOCK + wave) * 16;

    const int c = lane * 4;
    for (int r = 0; r < 16; ++r) {
        int g  = rowBase + r;
        int gg = (g < n2) ? g : 0;
        const float* selfp = h1 + (size_t)gg * 128;
        float4 sv = *(const float4*)(selfp + c);
        float ax = 0.f, ay = 0.f, az = 0.f, aw = 0.f;
        for (int s = 0; s < S_NBR; ++s) {
            int nb = nbr2[(size_t)gg * S_NBR + s];
            const float* fp = h1 + (size_t)nb * 128;
            float4 v = *(const float4*)(fp + c);
            ax += v.x; ay += v.y; az += v.z; aw += v.w;
        }
        float* xr = X + r * 256;
        xr[c + 0] = sv.x; xr[c + 1] = sv.y; xr[c + 2] = sv.z; xr[c + 3] = sv.w;
        xr[128 + c + 0] = ax * 0.1f; xr[128 + c + 1] = ay * 0.1f;
        xr[128 + c + 2] = az * 0.1f; xr[128 + c + 3] = aw * 0.1f;
    }
    __syncthreads();

    const int half = lane >> 4;
    const int lo   = lane & 15;
    const float* Xrow = X + lo * 256;

    v8f acc[4];
#pragma unroll
    for (int t = 0; t < 4; ++t) acc[t] = (v8f){0.f, 0.f, 0.f, 0.f, 0.f, 0.f, 0.f, 0.f};

    {
        const float* wp = Wself2 + (size_t)(2 * half) * 64 + lo;
        for (int k = 0; k < 32; ++k) {
            const int kk = 4 * k + 2 * half;
            v2f a;
            a.x = Xrow[kk];
            a.y = Xrow[kk + 1];
#pragma unroll
            for (int t = 0; t < 4; ++t) {
                v2f b;
                b.x = wp[t * 16];
                b.y = wp[64 + t * 16];
                acc[t] = __builtin_amdgcn_wmma_f32_16x16x4_f32(
                    false, a, false, b, (short)0, acc[t], false, false);
            }
            wp += 4 * 64;
        }
    }
    {
        const float* wp = Wnbr2 + (size_t)(2 * half) * 64 + lo;
        for (int k = 0; k < 32; ++k) {
            const int kk = 128 + 4 * k + 2 * half;
            v2f a;
            a.x = Xrow[kk];
            a.y = Xrow[kk + 1];
#pragma unroll
            for (int t = 0; t < 4; ++t) {
                v2f b;
                b.x = wp[t * 16];
                b.y = wp[64 + t * 16];
                acc[t] = __builtin_amdgcn_wmma_f32_16x16x4_f32(
                    false, a, false, b, (short)0, acc[t], false, false);
            }
            wp += 4 * 64;
        }
    }

#pragma unroll
    for (int t = 0; t < 4; ++t) {
        const int n = t * 16 + lo;
        const float bias = b2[n];
#pragma unroll
        for (int j = 0; j < 8; ++j) {
            int row = rowBase + 8 * half + j;
            if (row < n2) {
                h2[(size_t)row * 64 + n] = acc[t][j] + bias;
            }
        }
    }
}

// ---------------------------------------------------------------------------
// Cosine similarity per (source, item) pair. Masked sources reduce to sim = 0.
// rated_emb[src] = h2[inverse_all[n_items + src - n_masked]]  (src >= n_masked)
// item_emb row  = h2[inverse_all[item_rep_idx[i]]]
// ---------------------------------------------------------------------------
__global__ void cos_sim_kernel(const float* __restrict__ h2,
                               const int* __restrict__ inverse_all,
                               const int* __restrict__ source,
                               const int* __restrict__ item_rep_idx,
                               float* __restrict__ sim,
                               const int* __restrict__ n_items_p,
                               const int* __restrict__ n_masked_p,
                               int n_pairs) {
    int i = blockIdx.x * blockDim.x + threadIdx.x;
    if (i >= n_pairs) return;
    const int n_items  = *n_items_p;
    const int n_masked = *n_masked_p;
    int src = source[i];
    if (src < n_masked) {                                 // a == 0 -> sim = 0
        sim[i] = 0.f;
        return;
    }
    const float* arow = h2 + (size_t)inverse_all[n_items + src - n_masked] * 64;
    const float* brow = h2 + (size_t)inverse_all[item_rep_idx[i]] * 64;
    float dot = 0.f, na2 = 0.f, nb2 = 0.f;
    for (int col = 0; col < 64; col += 4) {
        float4 a = *(const float4*)(arow + col);
        float4 b = *(const float4*)(brow + col);
        dot += a.x * b.x + a.y * b.y + a.z * b.z + a.w * b.w;
        na2 += a.x * a.x + a.y * a.y + a.z * a.z + a.w * a.w;
        nb2 += b.x * b.x + b.y * b.y + b.z * b.z + b.w * b.w;
    }
    float na = fmaxf(sqrtf(na2), 1e-6f);
    float nb = fmaxf(sqrtf(nb2), 1e-6f);
    sim[i] = dot / (na * nb);
}

__global__ void reduce10_kernel(const float* __restrict__ sim,
                                float* __restrict__ pred,
                                int n_pred) {
    int j = blockIdx.x * blockDim.x + threadIdx.x;
    if (j >= n_pred) return;
    float s = 0.f;
    for (int k = 0; k < S_NBR; ++k) s += sim[j * S_NBR + k];
    pred[j] = s;
}

// ---------------------------------------------------------------------------
extern "C" void kernel_launch(void* const* d_in, const int* in_sizes, int n_in,
                              void* d_out, int out_size, void* d_ws, size_t ws_size,
                              hipStream_t stream) {
    const float* features     = (const float*)d_in[0];
    const float* Wself1       = (const float*)d_in[1];
    const float* Wnbr1        = (const float*)d_in[2];
    const float* b1           = (const float*)d_in[3];
    const float* Wself2       = (const float*)d_in[4];
    const float* Wnbr2        = (const float*)d_in[5];
    const float* b2           = (const float*)d_in[6];
    const int*   input_nodes  = (const int*)d_in[7];
    const int*   nbr1         = (const int*)d_in[8];
    const int*   nbr2         = (const int*)d_in[9];
    const int*   inverse_all  = (const int*)d_in[10];
    const int*   source       = (const int*)d_in[11];
    const int*   item_rep_idx = (const int*)d_in[12];
    const int*   n_items_p    = (const int*)d_in[13];
    const int*   n_masked_p   = (const int*)d_in[14];

    const int N1      = in_sizes[8] / S_NBR;              // 100000
    const int N2      = in_sizes[9] / S_NBR;              // 44096
    const int n_pairs = in_sizes[11];                     // 40960
    const int n_pred  = out_size;                         // 4096

    float* h1  = (float*)d_ws;
    float* h2  = h1 + (size_t)N1 * 128;
    float* sim = h2 + (size_t)N2 * 64;

    const size_t ldsBytes = (size_t)WAVES_PER_BLOCK * 16 * 256 * sizeof(float); // 64 KB

    const int rowsPerBlock = WAVES_PER_BLOCK * 16;
    dim3 blk(32 * WAVES_PER_BLOCK);

    sage_layer1<<<(N1 + rowsPerBlock - 1) / rowsPerBlock, blk, ldsBytes, stream>>>(
        features, Wself1, Wnbr1, b1, input_nodes, nbr1, h1, N1);

    sage_layer2<<<(N2 + rowsPerBlock - 1) / rowsPerBlock, blk, ldsBytes, stream>>>(
        h1, Wself2, Wnbr2, b2, nbr2, h2, N2);

    cos_sim_kernel<<<(n_pairs + 255) / 256, 256, 0, stream>>>(
        h2, inverse_all, source, item_rep_idx, sim, n_items_p, n_masked_p, n_pairs);

    reduce10_kernel<<<(n_pred + 255) / 256, 256, 0, stream>>>(sim, (float*)d_out, n_pred);
}